// GIN_16346645529220
// MI455X (gfx1250) — compile-verified
//
#include <hip/hip_runtime.h>

// ---------------------------------------------------------------------------
// GIN forward for MI455X (gfx1250, wave32).
// GEMMs via V_WMMA_F32_16X16X4_F32 with the A tile staged in LDS once per
// block (BN+relu fused into the stage); K and C are compile-time so all B
// loads / output stores use one base pointer + immediate offsets.
// Edge gather/scatter + pooling via f32 atomics (L2-resident: h = 25.6MB
// << 192MB L2).
// ---------------------------------------------------------------------------

typedef __attribute__((ext_vector_type(2))) float v2f;
typedef __attribute__((ext_vector_type(8))) float v8f;

#define NN   50000
#define EE   640000
#define DD   128
#define LL   4
#define GG   512
#define OUTD 10
#define BN_EPS 1e-5f

// h = atom_emb[nfeat]; agg = h (GIN eps=0 self term)
__global__ void k_embed(const int* __restrict__ nfeat, const float* __restrict__ aemb,
                        float* __restrict__ h, float* __restrict__ agg) {
    int i = blockIdx.x * blockDim.x + threadIdx.x;
    if (i >= NN * DD) return;
    int n = i >> 7, d = i & (DD - 1);
    float v = aemb[nfeat[n] * DD + d];
    h[i] = v;
    agg[i] = v;
}

// agg[dst] += relu(bond_emb[efeat] + h[src]) ; one thread per (edge, 4 floats)
__global__ void k_edge(const int* __restrict__ efeat, const int* __restrict__ src,
                       const int* __restrict__ dst, const float* __restrict__ bemb,
                       const float* __restrict__ h, float* __restrict__ agg) {
    int idx = blockIdx.x * blockDim.x + threadIdx.x;
    if (idx >= EE * (DD / 4)) return;
    int e = idx >> 5;              // DD/4 = 32 chunks per edge
    int c = (idx & 31) << 2;
    float4 he = *(const float4*)(bemb + efeat[e] * DD + c);
    float4 hs = *(const float4*)(h + (size_t)src[e] * DD + c);
    float* o = agg + (size_t)dst[e] * DD + c;
    atomicAdd(o + 0, fmaxf(he.x + hs.x, 0.f));
    atomicAdd(o + 1, fmaxf(he.y + hs.y, 0.f));
    atomicAdd(o + 2, fmaxf(he.z + hs.z, 0.f));
    atomicAdd(o + 3, fmaxf(he.w + hs.w, 0.f));
}

// Out[N,C] = act(A) @ W[K,C] + bias ; act = identity (BN=false) or relu(bn(.)) (BN=true).
// Block = 256 threads (8 waves) -> 16 rows x 128 cols; each wave owns one
// 16x16 tile via 16x16x4 fp32 WMMA. The 16xK A tile is staged in LDS once per
// block (BN+relu fused into the stage). Row pad of +4 floats keeps
// ds_load_2addr_b64 fragment reads bank-conflict-free (row r starts at bank 4r).
// Compile-time K,C => ds offsets, W loads and Out stores are immediate-offset.
template <bool BN, int K, int C>
__global__ void k_gemm(const float* __restrict__ A, const float* __restrict__ W,
                       const float* __restrict__ bias,
                       const float* __restrict__ scale, const float* __restrict__ shift,
                       float* __restrict__ Out) {
    constexpr int KP = K + 4;            // padded row stride (16B-aligned rows)
    __shared__ float As[16 * KP];

    const int row0 = blockIdx.x * 16;

    // cooperative stage: 16*K floats as float4 chunks, BN+relu applied once here
    for (int i = threadIdx.x; i < 16 * (K / 4); i += blockDim.x) {
        int r = i / (K / 4);
        int k = (i % (K / 4)) * 4;
        float4 x = *(const float4*)(A + (size_t)(row0 + r) * K + k);
        if (BN) {
            float4 sc = *(const float4*)(scale + k);
            float4 sh = *(const float4*)(shift + k);
            x.x = fmaxf(x.x * sc.x + sh.x, 0.f);
            x.y = fmaxf(x.y * sc.y + sh.y, 0.f);
            x.z = fmaxf(x.z * sc.z + sh.z, 0.f);
            x.w = fmaxf(x.w * sc.w + sh.w, 0.f);
        }
        *(float4*)(As + r * KP + k) = x;
    }
    __syncthreads();

    const int lane = threadIdx.x & 31;
    const int wave = threadIdx.x >> 5;
    const int half = lane >> 4;          // 0: K pair {0,1}, 1: K pair {2,3}
    const int l16  = lane & 15;
    const int col  = blockIdx.y * 128 + wave * 16 + l16;

    const float* arow  = As + l16 * KP + half * 2;        // A fragment base (LDS)
    const float* wbase = W + (size_t)(half * 2) * C + col; // B fragment base (global)

    v8f acc = {};
#pragma unroll 8
    for (int k0 = 0; k0 < K; k0 += 4) {
        v2f a = *(const v2f*)(arow + k0);      // ds_load b64, immediate offset
        v2f b;
        b.x = wbase[(size_t)k0 * C];           // immediate offsets off one base
        b.y = wbase[(size_t)k0 * C + C];
        acc = __builtin_amdgcn_wmma_f32_16x16x4_f32(false, a, false, b, 0, acc, false, false);
    }

    float bi = bias[col];
    float* obase = Out + (size_t)(row0 + half * 8) * C + col;
#pragma unroll
    for (int i = 0; i < 8; i++)
        obase[(size_t)i * C] = acc[i] + bi;    // immediate offsets off one base
}

__global__ void k_zero(float* __restrict__ p, int n) {
    int i = blockIdx.x * blockDim.x + threadIdx.x;
    if (i < n) p[i] = 0.f;
}

// per-column sum / sumsq over NN rows; blockDim.x == C, each block owns a row range
__global__ void k_bnstats(const float* __restrict__ X, float* __restrict__ sum,
                          float* __restrict__ sumsq, int C, int rows_per_block) {
    int c = threadIdx.x;
    int r0 = blockIdx.x * rows_per_block;
    int r1 = min(r0 + rows_per_block, NN);
    float s = 0.f, ss = 0.f;
    for (int r = r0; r < r1; r++) {
        float x = X[(size_t)r * C + c];
        s += x;
        ss += x * x;
    }
    atomicAdd(sum + c, s);
    atomicAdd(sumsq + c, ss);
}

// scale = g * rsqrt(var + eps), shift = b - mean * scale  (population var, as jnp.var)
__global__ void k_bnfin(const float* __restrict__ sum, const float* __restrict__ sumsq,
                        const float* __restrict__ g, const float* __restrict__ b,
                        float* __restrict__ scale, float* __restrict__ shift, int C) {
    int c = blockIdx.x * blockDim.x + threadIdx.x;
    if (c >= C) return;
    float m = sum[c] * (1.0f / NN);
    float v = sumsq[c] * (1.0f / NN) - m * m;
    float sc = g[c] * rsqrtf(v + BN_EPS);
    scale[c] = sc;
    shift[c] = b[c] - m * sc;
}

// h = act(bn2(t2)) + h ; agg = h  (prep next layer's self term)
__global__ void k_update(const float* __restrict__ t2, const float* __restrict__ scale,
                         const float* __restrict__ shift, float* __restrict__ h,
                         float* __restrict__ agg, int do_relu) {
    int i = blockIdx.x * blockDim.x + threadIdx.x;
    if (i >= NN * DD) return;
    int c = i & (DD - 1);
    float r = t2[i] * scale[c] + shift[c];
    if (do_relu) r = fmaxf(r, 0.f);
    float nh = r + h[i];
    h[i] = nh;
    agg[i] = nh;
}

// pooled[gid] += h[n] ; cnt[gid] += 1
__global__ void k_pool(const int* __restrict__ gid, const float* __restrict__ h,
                       float* __restrict__ pooled, float* __restrict__ cnt) {
    int idx = blockIdx.x * blockDim.x + threadIdx.x;
    if (idx >= NN * (DD / 4)) return;
    int n = idx >> 5;
    int c = (idx & 31) << 2;
    int g = gid[n];
    float4 v = *(const float4*)(h + (size_t)n * DD + c);
    float* o = pooled + g * DD + c;
    atomicAdd(o + 0, v.x);
    atomicAdd(o + 1, v.y);
    atomicAdd(o + 2, v.z);
    atomicAdd(o + 3, v.w);
    if (c == 0) atomicAdd(cnt + g, 1.0f);
}

// out[g,o] = (pooled[g]/max(cnt,1)) . Wp[:,o] + bp[o]   (512x10 — scalar is fine)
__global__ void k_proj(const float* __restrict__ pooled, const float* __restrict__ cnt,
                       const float* __restrict__ Wp, const float* __restrict__ bp,
                       float* __restrict__ out) {
    int idx = blockIdx.x * blockDim.x + threadIdx.x;
    if (idx >= GG * OUTD) return;
    int g = idx / OUTD, o = idx % OUTD;
    float inv = 1.0f / fmaxf(cnt[g], 1.0f);
    float acc = bp[o];
    for (int d = 0; d < DD; d++)
        acc += pooled[g * DD + d] * inv * Wp[d * OUTD + o];
    out[idx] = acc;
}

extern "C" void kernel_launch(void* const* d_in, const int* in_sizes, int n_in,
                              void* d_out, int out_size, void* d_ws, size_t ws_size,
                              hipStream_t stream) {
    const int*   nfeat = (const int*)d_in[0];
    const int*   efeat = (const int*)d_in[1];
    const int*   src   = (const int*)d_in[2];
    const int*   dst   = (const int*)d_in[3];
    const int*   gid   = (const int*)d_in[4];
    // d_in[5] = n_graphs scalar (compile-time GG)
    const float* aemb  = (const float*)d_in[6];
    const float* bemb  = (const float*)d_in[7];   // [L,5,D]
    const float* W1    = (const float*)d_in[8];   // [L,D,2D]
    const float* b1    = (const float*)d_in[9];   // [L,2D]
    const float* g1    = (const float*)d_in[10];
    const float* be1   = (const float*)d_in[11];
    const float* W2    = (const float*)d_in[12];  // [L,2D,D]
    const float* b2    = (const float*)d_in[13];  // [L,D]
    const float* g2    = (const float*)d_in[14];
    const float* be2   = (const float*)d_in[15];
    const float* Wp    = (const float*)d_in[16];  // [D,OUT]
    const float* bp    = (const float*)d_in[17];
    float* out = (float*)d_out;

    // workspace layout (floats)
    float* ws     = (float*)d_ws;
    float* h      = ws;                       // N*D
    float* agg    = h + (size_t)NN * DD;      // N*D
    float* t      = agg + (size_t)NN * DD;    // N*2D
    float* t2     = t + (size_t)NN * 2 * DD;  // N*D
    float* bsum   = t2 + (size_t)NN * DD;     // 256
    float* bsq    = bsum + 256;               // 256 (contiguous with bsum)
    float* bsc    = bsq + 256;                // 256
    float* bsh    = bsc + 256;                // 256
    float* pooled = bsh + 256;                // G*D
    float* cnt    = pooled + (size_t)GG * DD; // G (contiguous with pooled)

    const int T = 256;
    k_embed<<<(NN * DD + T - 1) / T, T, 0, stream>>>(nfeat, aemb, h, agg);

    for (int l = 0; l < LL; l++) {
        // message passing: agg = h + segment_sum(relu(be + h[src]), dst)
        k_edge<<<(EE * (DD / 4) + T - 1) / T, T, 0, stream>>>(
            efeat, src, dst, bemb + (size_t)l * 5 * DD, h, agg);

        // t = agg @ W1 + b1   [N,256]
        dim3 grid1(NN / 16, (2 * DD) / 128);
        k_gemm<false, DD, 2 * DD><<<grid1, T, 0, stream>>>(
            agg, W1 + (size_t)l * DD * 2 * DD, b1 + (size_t)l * 2 * DD,
            nullptr, nullptr, t);

        // BN1 stats + finalize
        k_zero<<<2, T, 0, stream>>>(bsum, 512);
        k_bnstats<<<250, 2 * DD, 0, stream>>>(t, bsum, bsq, 2 * DD, 200);
        k_bnfin<<<1, T, 0, stream>>>(bsum, bsq, g1 + (size_t)l * 2 * DD,
                                     be1 + (size_t)l * 2 * DD, bsc, bsh, 2 * DD);

        // t2 = relu(bn1(t)) @ W2 + b2   [N,128]  (BN+relu fused into LDS stage)
        dim3 grid2(NN / 16, DD / 128);
        k_gemm<true, 2 * DD, DD><<<grid2, T, 0, stream>>>(
            t, W2 + (size_t)l * 2 * DD * DD, b2 + (size_t)l * DD,
            bsc, bsh, t2);

        // BN2 stats + finalize
        k_zero<<<2, T, 0, stream>>>(bsum, 512);
        k_bnstats<<<250, DD, 0, stream>>>(t2, bsum, bsq, DD, 200);
        k_bnfin<<<1, T, 0, stream>>>(bsum, bsq, g2 + (size_t)l * DD,
                                     be2 + (size_t)l * DD, bsc, bsh, DD);

        // h = act(bn2(t2)) + h ; agg = h
        k_update<<<(NN * DD + T - 1) / T, T, 0, stream>>>(
            t2, bsc, bsh, h, agg, (l != LL - 1) ? 1 : 0);
    }

    // graph mean pool + projection
    k_zero<<<(GG * DD + GG + T - 1) / T, T, 0, stream>>>(pooled, GG * DD + GG);
    k_pool<<<(NN * (DD / 4) + T - 1) / T, T, 0, stream>>>(gid, h, pooled, cnt);
    k_proj<<<(GG * OUTD + T - 1) / T, T, 0, stream>>>(pooled, cnt, Wp, bp, out);
}